// ScaledDotProductAttention_46308337386102
// MI455X (gfx1250) — compile-verified
//
#include <hip/hip_runtime.h>
#include <hip/hip_bf16.h>

typedef __attribute__((ext_vector_type(2))) float v2f;
typedef __attribute__((ext_vector_type(8))) float v8f;

#define TEMP_INV 0.125f   // 1/sqrt(64)
#define LQ 512
#define LK 512
#define DD 64
#define QTILE 16

// One block = (batch b, 16-query tile). 4 waves * 32 lanes.
// Wave w owns key columns [w*128, w*128+128) for scores, and output
// features [w*16, w*16+16) for P@V.
__global__ __launch_bounds__(128)
void attn_f32wmma_kernel(const float* __restrict__ q,
                         const float* __restrict__ k,
                         const float* __restrict__ v,
                         const unsigned char* __restrict__ key_mask,
                         const unsigned char* __restrict__ query_mask,
                         float* __restrict__ out,
                         float* __restrict__ attn) {
  __shared__ float Qs[QTILE * 65];        // padded: stride 65 -> no bank conflicts
  __shared__ float Ks[4 * QTILE * 65];    // per-wave K tile staging (async DMA dest)
  __shared__ float Ss[QTILE * 513];       // scores / probs, padded stride 513
  __shared__ float Red[QTILE * 8];
  __shared__ float Rmax[QTILE];
  __shared__ float Rsum[QTILE];

  const int tid  = threadIdx.x;
  const int wave = tid >> 5;
  const int lane = tid & 31;
  const int b     = blockIdx.x >> 5;      // 32 q-tiles per batch
  const int qt    = blockIdx.x & 31;
  const int qbase = qt * QTILE;

  const float* qB = q + ((size_t)b * LQ + qbase) * DD;
  const float* kB = k + (size_t)b * LK * DD;
  const float* vB = v + (size_t)b * LK * DD;
  const unsigned char* kmB = key_mask   + ((size_t)b * LQ + qbase) * LK;
  const unsigned char* qmB = query_mask + ((size_t)b * LQ + qbase) * LK;
  float* outB  = out  + ((size_t)b * LQ + qbase) * DD;
  float* attnB = attn + ((size_t)b * LQ + qbase) * LK;

  // ---- stage Q tile (coalesced) ----
  for (int i = tid; i < QTILE * DD; i += 128) {
    int r = i >> 6, c = i & 63;
    Qs[r * 65 + c] = qB[r * DD + c];
  }
  __syncthreads();

  // WMMA f32 16x16x4 A-layout: lanes 0-15 row M=lane hold K={0,1};
  // lanes 16-31 row M=lane-16 hold K={2,3}.
  const int row  = lane & 15;
  const int koff = (lane >> 4) * 2;

  // Q A-fragments for the 16 chained K=4 steps covering D=64.
  v2f qa[16];
#pragma unroll
  for (int kk = 0; kk < 16; ++kk) {
    qa[kk].x = Qs[row * 65 + 4 * kk + koff];
    qa[kk].y = Qs[row * 65 + 4 * kk + koff + 1];
  }

  float* Kw = Ks + wave * QTILE * 65;
  // Low 32 bits of a flat LDS pointer = LDS byte offset (ISA aperture map),
  // which is what GLOBAL_LOAD_ASYNC_TO_LDS expects in VDST.
  const unsigned KwOff = (unsigned)(size_t)(void*)Kw;

  // ---- scores: S = (Q K^T) / T, key-mask -> -inf, into LDS ----
  // K tile is wave-private: stage it with CDNA5 async global->LDS DMA
  // (ASYNCcnt) instead of block barriers.
  for (int ct = 0; ct < 8; ++ct) {
    const int colbase = wave * 128 + ct * 16;

    // WAR guard: previous tile's ds_load fragments must be complete
    // before the DMA engine overwrites the buffer.
    asm volatile("s_wait_dscnt 0x0" ::: "memory");

    // 16 rows x 64 floats = 256 x 16B chunks; 8 b128 DMAs per lane.
#pragma unroll
    for (int it = 0; it < 8; ++it) {
      const int id = lane + 32 * it;           // chunk id 0..255
      const int r  = id >> 4;                  // key row in tile
      const int cc = (id & 15) * 4;            // feature (dwords)
      const unsigned lds_addr = KwOff + (unsigned)(r * 65 + cc) * 4u;
      const unsigned goff = (unsigned)(((colbase + r) * DD + cc) * 4);
      asm volatile("global_load_async_to_lds_b128 %0, %1, %2"
                   :: "v"(lds_addr), "v"(goff), "s"(kB)
                   : "memory");
    }
    asm volatile("s_wait_asynccnt 0x0" ::: "memory");

    v8f acc = {};
#pragma unroll
    for (int kk = 0; kk < 16; ++kk) {
      // B-layout (4x16): lanes 0-15 col N=lane hold K-rows {0,1};
      // lanes 16-31 hold K-rows {2,3}. B[kr][n] = K[coltile*16+n][4kk+kr].
      v2f bf;
      bf.x = Kw[(lane & 15) * 65 + 4 * kk + koff];
      bf.y = Kw[(lane & 15) * 65 + 4 * kk + koff + 1];
      acc = __builtin_amdgcn_wmma_f32_16x16x4_f32(
          false, qa[kk], false, bf, (short)0, acc, false, false);
    }

    // C layout: vgpr vv, lanes 0-15: M=vv; lanes 16-31: M=vv+8; N=lane&15
#pragma unroll
    for (int vv = 0; vv < 8; ++vv) {
      int M   = vv + ((lane >> 4) << 3);
      int col = colbase + (lane & 15);
      bool km = kmB[(size_t)M * LK + col] != 0;
      Ss[M * 513 + col] = km ? -__builtin_inff() : acc[vv] * TEMP_INV;
    }
  }
  __syncthreads();

  // ---- softmax over each 512-wide row (segmented reduction) ----
  const int rrow = tid >> 3;   // 16 rows x 8 segments of 64
  const int seg  = tid & 7;
  float mx = -__builtin_inff();
  for (int j = 0; j < 64; ++j) mx = fmaxf(mx, Ss[rrow * 513 + seg * 64 + j]);
  Red[rrow * 8 + seg] = mx;
  __syncthreads();
  if (seg == 0) {
    float m2 = Red[rrow * 8];
    for (int j = 1; j < 8; ++j) m2 = fmaxf(m2, Red[rrow * 8 + j]);
    Rmax[rrow] = m2;
  }
  __syncthreads();
  float rm = Rmax[rrow];
  float sm = 0.f;
  for (int j = 0; j < 64; ++j) sm += __expf(Ss[rrow * 513 + seg * 64 + j] - rm);
  Red[rrow * 8 + seg] = sm;
  __syncthreads();
  if (seg == 0) {
    float s2 = 0.f;
    for (int j = 0; j < 8; ++j) s2 += Red[rrow * 8 + j];
    Rsum[rrow] = s2;
  }
  __syncthreads();

  // ---- normalize, post-softmax query-mask zero, write attn (coalesced) ----
  for (int idx = tid; idx < QTILE * LK; idx += 128) {
    int r = idx >> 9, c = idx & 511;
    float p = __expf(Ss[r * 513 + c] - Rmax[r]) / Rsum[r];
    if (qmB[(size_t)r * LK + c]) p = 0.f;
    Ss[r * 513 + c] = p;
    attnB[(size_t)r * LK + c] = p;
  }
  __syncthreads();

  // ---- O = P @ V : wave w computes output features [w*16, w*16+16) ----
  v8f o = {};
  const int n = lane & 15;
  for (int kk = 0; kk < 128; ++kk) {
    v2f a, bf;
    a.x = Ss[row * 513 + 4 * kk + koff];
    a.y = Ss[row * 513 + 4 * kk + koff + 1];
    const float* vp = vB + (size_t)(4 * kk + koff) * DD + wave * 16 + n;
    bf.x = vp[0];
    bf.y = vp[DD];
    o = __builtin_amdgcn_wmma_f32_16x16x4_f32(
        false, a, false, bf, (short)0, o, false, false);
  }
#pragma unroll
  for (int vv = 0; vv < 8; ++vv) {
    int M = vv + ((lane >> 4) << 3);
    outB[(size_t)M * DD + wave * 16 + n] = o[vv];
  }
}

extern "C" void kernel_launch(void* const* d_in, const int* in_sizes, int n_in,
                              void* d_out, int out_size, void* d_ws, size_t ws_size,
                              hipStream_t stream) {
  (void)in_sizes; (void)n_in; (void)out_size; (void)d_ws; (void)ws_size;
  const float* q = (const float*)d_in[0];
  const float* k = (const float*)d_in[1];
  const float* v = (const float*)d_in[2];
  const unsigned char* key_mask   = (const unsigned char*)d_in[3];
  const unsigned char* query_mask = (const unsigned char*)d_in[4];

  float* out  = (float*)d_out;                       // (128,512,64)
  float* attn = out + (size_t)128 * 512 * 64;        // (128,512,512)

  dim3 grid(128 * (LQ / QTILE));   // batch * q-tiles = 4096
  dim3 block(128);                 // 4 waves (wave32)
  attn_f32wmma_kernel<<<grid, block, 0, stream>>>(q, k, v, key_mask, query_mask,
                                                  out, attn);
}